// VNL_Loss_6691559047750
// MI455X (gfx1250) — compile-verified
//
#include <hip/hip_runtime.h>

#define HH 1080
#define WW 1920
#define HWPIX (HH * WW)
#define NB 4096
#define LMAXF 6.0f
#define D_COS 0.867f
#define D_DIFF 0.35f
#define D_Z 0.05f
#define D_FARZ 12.3f

typedef float v2f __attribute__((ext_vector_type(2)));
typedef float v8f __attribute__((ext_vector_type(8)));

struct F3 { float x, y, z; };

__device__ __forceinline__ F3 f3sub(F3 a, F3 b) { return {a.x - b.x, a.y - b.y, a.z - b.z}; }
__device__ __forceinline__ F3 f3cross(F3 a, F3 b) {
  return {a.y * b.z - a.z * b.y, a.z * b.x - a.x * b.z, a.x * b.y - a.y * b.x};
}
__device__ __forceinline__ F3 unit_normal(F3 a, F3 b) {
  F3 c = f3cross(a, b);
  float nn = sqrtf(c.x * c.x + c.y * c.y + c.z * c.z);
  nn += (nn == 0.0f) ? 0.01f : 0.0f;
  float inv = 1.0f / nn;
  return {c.x * inv, c.y * inv, c.z * inv};
}
__device__ __forceinline__ F3 mkpoint(int idx, float z) {
  float u = (float)(idx % WW - WW / 2);
  float v = (float)(idx / WW - HH / 2);
  float ad = fabsf(z);
  return {u * ad * 1.0e-3f, v * ad * 1.0e-3f, z};  // FX = FY = 1000
}
__device__ __forceinline__ int cos_hit(float e, float nm) {
  float ne = e / (nm + 1e-8f);
  return ((ne > D_COS) || (ne < -D_COS)) ? 1 : 0;
}

// ---------------------------------------------------------------------------
// Pass 1: gather points, WMMA Gram (einsum bgci,bgcj->bgij), masks, loss,
//         per-block LDS histogram of masked losses.
// ---------------------------------------------------------------------------
__global__ __launch_bounds__(256) void vnl_main(
    const float* __restrict__ pred, const float* __restrict__ gt,
    const int* __restrict__ p1, const int* __restrict__ p2,
    const int* __restrict__ p3, int Gn, int Ntot,
    float* __restrict__ larr, unsigned* __restrict__ hcnt,
    float* __restrict__ hsum) {
  __shared__ float vecs[8][97][4];   // per-wave edge vectors; slot 96 = zeros
  __shared__ float dbuf[8][256];     // per-wave WMMA D tile (16x16)
  __shared__ unsigned s_cnt[NB];
  __shared__ float s_sum[NB];

  const int tid = threadIdx.x;
  for (int ii = tid; ii < NB; ii += 256) { s_cnt[ii] = 0u; s_sum[ii] = 0.0f; }
  __syncthreads();

  const int lane = tid & 31;
  const int wave = tid >> 5;
  const int gid = blockIdx.x * 256 + tid;
  const bool active = (gid < Ntot);
  const int cg = active ? gid : (Ntot - 1);   // clamp: keep EXEC all-ones for WMMA
  const int b = cg / Gn;
  const int ig = cg - b * Gn;

  const int i1 = p1[ig], i2 = p2[ig], i3 = p3[ig];
  const float* gdep = gt + (size_t)b * HWPIX;
  const float* pdep = pred + (size_t)b * HWPIX;
  const float gz1 = gdep[i1], gz2 = gdep[i2], gz3 = gdep[i3];
  const float qz1 = pdep[i1], qz2 = pdep[i2], qz3 = pdep[i3];

  F3 G1 = mkpoint(i1, gz1), G2 = mkpoint(i2, gz2), G3 = mkpoint(i3, gz3);
  F3 P1 = mkpoint(i1, qz1), P2 = mkpoint(i2, qz2), P3 = mkpoint(i3, qz3);

  // replicate reference's z_zero broadcast: row-c of all points -> 1e-4
  if (P1.z == 0.0f) { P1.x = P2.x = P3.x = 1.0e-4f; }
  if (P2.z == 0.0f) { P1.y = P2.y = P3.y = 1.0e-4f; }
  if (P3.z == 0.0f) { P1.z = P2.z = P3.z = 1.0e-4f; }

  const F3 d12 = f3sub(G2, G1), d13 = f3sub(G3, G1), d23 = f3sub(G3, G2);

  // stage this lane's 3 edge vectors for the wave's WMMA tiles
  float(*vv)[4] = vecs[wave];
  if (lane == 0) { vv[96][0] = 0.f; vv[96][1] = 0.f; vv[96][2] = 0.f; vv[96][3] = 0.f; }
  vv[3 * lane + 0][0] = d12.x; vv[3 * lane + 0][1] = d12.y;
  vv[3 * lane + 0][2] = d12.z; vv[3 * lane + 0][3] = 0.0f;
  vv[3 * lane + 1][0] = d13.x; vv[3 * lane + 1][1] = d13.y;
  vv[3 * lane + 1][2] = d13.z; vv[3 * lane + 1][3] = 0.0f;
  vv[3 * lane + 2][0] = d23.x; vv[3 * lane + 2][1] = d23.y;
  vv[3 * lane + 2][2] = d23.z; vv[3 * lane + 2][3] = 0.0f;

  // Gram matrices via V_WMMA_F32_16X16X4_F32: 7 tiles x 5 triangles.
  // A(16x4) rows = edge vectors; D = A * A^T (B fragment == A fragment).
  // Branch-free fragment build: invalid slots alias zeroed slot 96;
  // lanes 0-15 read the (x,y) pair (K0/K1), lanes 16-31 the (z,0) pair (K2/K3).
  float e00 = 0.f, e11 = 0.f, e22 = 0.f, e01 = 0.f, e02 = 0.f, e12 = 0.f;
  const int tq = lane / 5;                // this lane's tile
  const int rr = lane - tq * 5;           // triangle slot within tile
  const int s = lane & 15;
  const int comp = (lane >> 4) << 1;      // 0 or 2
  float* db = dbuf[wave];
  for (int t = 0; t < 7; ++t) {
    const int vidx = 15 * t + s;
    const bool valid = (s < 15) && (vidx < 96);
    const int slot = valid ? vidx : 96;
    const v2f a = *(const v2f*)&vv[slot][comp];   // single ds_load_b64
    v8f c = {};
    v8f d = __builtin_amdgcn_wmma_f32_16x16x4_f32(false, a, false, a,
                                                  (short)0, c, false, false);
    const int row0 = (lane < 16) ? 0 : 8;
#pragma unroll
    for (int r8 = 0; r8 < 8; ++r8) db[(row0 + r8) * 16 + s] = d[r8];
    if (t == tq) {  // lanes owning this tile extract their 3x3 Gram block
      const int bs = 3 * rr;
      e00 = db[(bs + 0) * 16 + bs + 0];
      e11 = db[(bs + 1) * 16 + bs + 1];
      e22 = db[(bs + 2) * 16 + bs + 2];
      e01 = db[(bs + 0) * 16 + bs + 1];
      e02 = db[(bs + 0) * 16 + bs + 2];
      e12 = db[(bs + 1) * 16 + bs + 2];
    }
  }

  // masks
  const float n0 = sqrtf(e00), n1 = sqrtf(e11), n2 = sqrtf(e22);
  const int chit = cos_hit(e00, n0 * n0) + cos_hit(e11, n1 * n1) +
                   cos_hit(e22, n2 * n2) +
                   2 * (cos_hit(e01, n0 * n1) + cos_hit(e02, n0 * n2) +
                        cos_hit(e12, n1 * n2));
  const bool mask_cos = chit > 3;
  const bool mask_x = (fabsf(d12.x) < D_DIFF) || (fabsf(d13.x) < D_DIFF) || (fabsf(d23.x) < D_DIFF);
  const bool mask_y = (fabsf(d12.y) < D_DIFF) || (fabsf(d13.y) < D_DIFF) || (fabsf(d23.y) < D_DIFF);
  const bool mask_z = (fabsf(d12.z) < D_DIFF) || (fabsf(d13.z) < D_DIFF) || (fabsf(d23.z) < D_DIFF);
  const bool mask_pad = (gz1 > D_Z) && (gz2 > D_Z) && (gz3 > D_Z);
  const bool mask_far = (gz1 < D_FARZ) && (gz2 < D_FARZ) && (gz3 < D_FARZ);
  const bool mask_near = !((mask_x && mask_y && mask_z) || mask_cos);
  const bool m = mask_pad && mask_near && mask_far;

  // normals + loss
  const F3 gn = unit_normal(d12, d13);
  const F3 dn = unit_normal(f3sub(P2, P1), f3sub(P3, P1));
  const float loss = fabsf(gn.x - dn.x) + fabsf(gn.y - dn.y) + fabsf(gn.z - dn.z);

  if (active) {
    larr[gid] = m ? loss : -__builtin_inff();
    if (m) {
      int bin = (int)(loss * ((float)NB / LMAXF));
      bin = bin < 0 ? 0 : (bin > NB - 1 ? NB - 1 : bin);
      atomicAdd(&s_cnt[bin], 1u);
      atomicAdd(&s_sum[bin], loss);
    }
  }
  __syncthreads();
  for (int ii = tid; ii < NB; ii += 256) {
    if (s_cnt[ii]) {
      atomicAdd(&hcnt[ii], s_cnt[ii]);
      atomicAdd(&hsum[ii], s_sum[ii]);
    }
  }
}

// ---------------------------------------------------------------------------
// Pass 2: coarse scan — total count/sum, cutoff bin for dropping V/4 smallest.
// ---------------------------------------------------------------------------
__global__ void vnl_scan1(const unsigned* __restrict__ hcnt,
                          const float* __restrict__ hsum,
                          unsigned* __restrict__ sc) {
  if (threadIdx.x != 0 || blockIdx.x != 0) return;
  unsigned V = 0; float S = 0.0f;
  for (int i = 0; i < NB; ++i) { V += hcnt[i]; S += hsum[i]; }
  const unsigned k = V / 4u;  // drop count
  int b1 = -1; unsigned krem = 0, c = 0; float sb = 0.0f;
  if (k > 0) {
    for (int i = 0; i < NB; ++i) {
      const unsigned ci = hcnt[i];
      if (c + ci <= k) { c += ci; sb += hsum[i]; if (c == k) break; }
      else { b1 = i; krem = k - c; break; }
    }
  }
  sc[0] = V;
  ((int*)sc)[1] = b1;
  sc[2] = krem;
  ((float*)sc)[3] = S;
  ((float*)sc)[4] = sb;
}

// ---------------------------------------------------------------------------
// Pass 3: fine histogram inside cutoff bin.
// ---------------------------------------------------------------------------
__global__ __launch_bounds__(256) void vnl_fine(const float* __restrict__ larr,
                                                const unsigned* __restrict__ sc,
                                                int Ntot,
                                                unsigned* __restrict__ h2cnt,
                                                float* __restrict__ h2sum) {
  const int gid = blockIdx.x * 256 + threadIdx.x;
  if (gid >= Ntot) return;
  const int b1 = ((const int*)sc)[1];
  if (b1 < 0) return;
  const float l = larr[gid];
  if (!(l >= 0.0f)) return;  // rejects -inf (unmasked)
  int bin = (int)(l * ((float)NB / LMAXF));
  bin = bin < 0 ? 0 : (bin > NB - 1 ? NB - 1 : bin);
  if (bin != b1) return;
  const float lo = (float)b1 * (LMAXF / (float)NB);
  int fb = (int)((l - lo) * ((float)NB * (float)NB / LMAXF));
  fb = fb < 0 ? 0 : (fb > NB - 1 ? NB - 1 : fb);
  atomicAdd(&h2cnt[fb], 1u);
  atomicAdd(&h2sum[fb], l);
}

// ---------------------------------------------------------------------------
// Pass 4: final combine.
// ---------------------------------------------------------------------------
__global__ void vnl_final(const unsigned* __restrict__ sc,
                          const unsigned* __restrict__ h2cnt,
                          const float* __restrict__ h2sum,
                          float* __restrict__ out) {
  if (threadIdx.x != 0 || blockIdx.x != 0) return;
  const unsigned V = sc[0];
  const int b1 = ((const int*)sc)[1];
  const unsigned krem = sc[2];
  const float S = ((const float*)sc)[3];
  float dropped = ((const float*)sc)[4];
  if (b1 >= 0 && krem > 0) {
    unsigned c = 0;
    for (int f = 0; f < NB; ++f) {
      const unsigned cf = h2cnt[f];
      if (cf == 0) continue;
      if (c + cf <= krem) { dropped += h2sum[f]; c += cf; if (c == krem) break; }
      else { const unsigned take = krem - c; dropped += h2sum[f] * ((float)take / (float)cf); break; }
    }
  }
  const unsigned keep = V - V / 4u;
  const float denom = (keep > 0u) ? (float)keep : 1.0f;
  out[0] = (S - dropped) / denom;
}

// ---------------------------------------------------------------------------
extern "C" void kernel_launch(void* const* d_in, const int* in_sizes, int n_in,
                              void* d_out, int out_size, void* d_ws,
                              size_t ws_size, hipStream_t stream) {
  const float* pred = (const float*)d_in[0];
  const float* gt = (const float*)d_in[1];
  const int* p1 = (const int*)d_in[2];
  const int* p2 = (const int*)d_in[3];
  const int* p3 = (const int*)d_in[4];
  const int Gn = in_sizes[2];
  const int Bn = in_sizes[0] / HWPIX;
  const int Ntot = Bn * Gn;

  float* larr = (float*)d_ws;
  char* base = (char*)d_ws + (size_t)Ntot * sizeof(float);
  unsigned* hcnt = (unsigned*)base;
  float* hsum = (float*)(base + NB * 4);
  unsigned* h2cnt = (unsigned*)(base + NB * 8);
  float* h2sum = (float*)(base + NB * 12);
  unsigned* sc = (unsigned*)(base + NB * 16);

  hipMemsetAsync(base, 0, NB * 16 + 64, stream);

  const int blocks = (Ntot + 255) / 256;
  vnl_main<<<blocks, 256, 0, stream>>>(pred, gt, p1, p2, p3, Gn, Ntot, larr,
                                       hcnt, hsum);
  vnl_scan1<<<1, 1, 0, stream>>>(hcnt, hsum, sc);
  vnl_fine<<<blocks, 256, 0, stream>>>(larr, sc, Ntot, h2cnt, h2sum);
  vnl_final<<<1, 1, 0, stream>>>(sc, h2cnt, h2sum, (float*)d_out);
}